// SA1_43843026157963
// MI455X (gfx1250) — compile-verified
//
#include <hip/hip_runtime.h>
#include <hip/hip_bf16.h>

typedef float v2f __attribute__((ext_vector_type(2)));
typedef float v8f __attribute__((ext_vector_type(8)));

#define NB   4
#define CIN  256
#define NPIX 4096
#define DH   64

__device__ __forceinline__ v8f wmma_f32x4(v8f acc, v2f a, v2f b) {
  // D = A(16x4) * B(4x16) + C, fp32 throughout
  return __builtin_amdgcn_wmma_f32_16x16x4_f32(false, a, false, b, (short)0, acc, false, false);
}

#define LANE_IDS()                         \
  const int lane = threadIdx.x & 31;       \
  const int half = lane >> 4;              \
  const int mn   = lane & 15;              \
  const int wave = threadIdx.x >> 5;       \
  (void)wave;

// ---------------- 1. QKV projection: [64,256] x [256,4096] per (b, which) ----
__global__ __launch_bounds__(128) void k_qkv(
    const float* __restrict__ x,
    const float* __restrict__ wq, const float* __restrict__ bq,
    const float* __restrict__ wk, const float* __restrict__ bk,
    const float* __restrict__ wv, const float* __restrict__ bv,
    float* __restrict__ q, float* __restrict__ k, float* __restrict__ v)
{
  LANE_IDS();
  const int n0    = (blockIdx.x * 4 + wave) * 16;
  const int which = blockIdx.y >> 2;
  const int m0    = (blockIdx.y & 3) * 16;
  const int b     = blockIdx.z;

  const float* W    = (which == 0) ? wq : (which == 1) ? wk : wv;
  const float* bias = (which == 0) ? bq : (which == 1) ? bk : bv;
  float*       O    = (which == 0) ? q  : (which == 1) ? k  : v;

  const float* ap = W + (size_t)(m0 + mn) * CIN + 2 * half;              // A: row m, K pair
  const float* bp = x + (size_t)b * CIN * NPIX + (size_t)(2 * half) * NPIX + n0 + mn;

  v8f acc = {};
  for (int kk = 0; kk < CIN; kk += 4) {
    v2f a;  a.x  = ap[kk];                    a.y  = ap[kk + 1];
    v2f bb; bb.x = bp[(size_t)kk * NPIX];     bb.y = bp[(size_t)(kk + 1) * NPIX];
    acc = wmma_f32x4(acc, a, bb);
  }
  float* ob = O + (size_t)b * DH * NPIX;
#pragma unroll
  for (int r = 0; r < 8; ++r) {
    const int m = m0 + r + 8 * half;
    ob[(size_t)m * NPIX + n0 + mn] = acc[r] + bias[m];
  }
}

// ---------------- 2. scores[b,i,j] = sum_d q[d,i]*k[d,j]  (K=64) -------------
__global__ __launch_bounds__(128) void k_scores(
    const float* __restrict__ q, const float* __restrict__ k,
    float* __restrict__ s)
{
  LANE_IDS();
  const int j0 = (blockIdx.x * 4 + wave) * 16;
  const int i0 = blockIdx.y * 16;
  const int b  = blockIdx.z;

  const float* qb = q + (size_t)b * DH * NPIX + i0 + mn;   // A = q^T (coalesced over i)
  const float* kb = k + (size_t)b * DH * NPIX + j0 + mn;   // B = k   (coalesced over j)

  v8f acc = {};
  for (int d = 0; d < DH; d += 4) {
    const int row = d + 2 * half;
    v2f a;  a.x  = qb[(size_t)row * NPIX];  a.y  = qb[(size_t)(row + 1) * NPIX];
    v2f bb; bb.x = kb[(size_t)row * NPIX];  bb.y = kb[(size_t)(row + 1) * NPIX];
    acc = wmma_f32x4(acc, a, bb);
  }
  float* sb = s + ((size_t)b << 24);
#pragma unroll
  for (int r = 0; r < 8; ++r)
    sb[(size_t)(i0 + r + 8 * half) * NPIX + j0 + mn] = acc[r];
}

// ---------------- 3. softmax over axis=1 (over i, per column j), in place ----
__global__ __launch_bounds__(256) void k_softmax(float* __restrict__ s) {
  const int col = blockIdx.x * 256 + threadIdx.x;  // 0..16383
  const int b = col >> 12;
  const int j = col & 4095;
  float* base = s + ((size_t)b << 24) + j;         // column stride = 4096 floats

  float m = -3.4e38f;
  for (int i = 0; i < 4096; ++i) m = fmaxf(m, base[(size_t)i << 12]);
  float sum = 0.f;
  for (int i = 0; i < 4096; ++i) sum += __expf(base[(size_t)i << 12] - m);
  const float inv = 1.0f / sum;
  for (int i = 0; i < 4096; ++i) {
    const size_t o = (size_t)i << 12;
    base[o] = __expf(base[o] - m) * inv;
  }
}

// ---------------- 4. vT[b][j][d] = v[b][d][j] --------------------------------
__global__ __launch_bounds__(256) void k_transpose_v(
    const float* __restrict__ v, float* __restrict__ vT)
{
  const size_t idx = (size_t)blockIdx.x * 256 + threadIdx.x;  // 4*64*4096
  const int b = (int)(idx >> 18);
  const int d = (int)((idx >> 12) & 63);
  const int j = (int)(idx & 4095);
  vT[((size_t)b << 18) + (size_t)j * DH + d] = v[idx];
}

// ---------------- 5. x_hat[b,i,d] = sum_j alpha[i,j]*vT[j,d]  (K=4096) -------
__global__ __launch_bounds__(128) void k_xhat(
    const float* __restrict__ alpha, const float* __restrict__ vT,
    float* __restrict__ xh)
{
  LANE_IDS();
  const int d0 = wave * 16;           // 4 waves cover D=64
  const int i0 = blockIdx.x * 16;
  const int b  = blockIdx.y;

  const float* al = alpha + ((size_t)b << 24) + (size_t)(i0 + mn) * NPIX + 2 * half;
  const float* vb = vT    + ((size_t)b << 18) + (size_t)(2 * half) * DH + d0 + mn;

  v8f acc = {};
  for (int j = 0; j < NPIX; j += 4) {
    v2f a;  a.x  = al[j];                  a.y  = al[j + 1];
    v2f bb; bb.x = vb[(size_t)j * DH];     bb.y = vb[(size_t)(j + 1) * DH];
    acc = wmma_f32x4(acc, a, bb);
  }
  float* ob = xh + ((size_t)b << 18);     // contiguous [B,N,D] == raw view [B,64,H,W]
#pragma unroll
  for (int r = 0; r < 8; ++r)
    ob[(size_t)(i0 + r + 8 * half) * DH + d0 + mn] = acc[r];
}

// ---------------- 6. y = w_out[256,64] x xh_view[64,4096] + b_out ------------
__global__ __launch_bounds__(128) void k_ygemm(
    const float* __restrict__ xh, const float* __restrict__ wout,
    const float* __restrict__ bout, float* __restrict__ y)
{
  LANE_IDS();
  const int p0 = (blockIdx.x * 4 + wave) * 16;
  const int c0 = blockIdx.y * 16;
  const int b  = blockIdx.z;

  const float* ap = wout + (size_t)(c0 + mn) * DH + 2 * half;
  const float* bp = xh + ((size_t)b << 18) + (size_t)(2 * half) * NPIX + p0 + mn;

  v8f acc = {};
  for (int kk = 0; kk < DH; kk += 4) {
    v2f a;  a.x  = ap[kk];                 a.y  = ap[kk + 1];
    v2f bb; bb.x = bp[(size_t)kk * NPIX];  bb.y = bp[(size_t)(kk + 1) * NPIX];
    acc = wmma_f32x4(acc, a, bb);
  }
  float* ob = y + (size_t)b * CIN * NPIX;
#pragma unroll
  for (int r = 0; r < 8; ++r) {
    const int c = c0 + r + 8 * half;
    ob[(size_t)c * NPIX + p0 + mn] = acc[r] + bout[c];
  }
}

// ---------------- 7. BN stats: per-channel sum and sum of squares ------------
__global__ __launch_bounds__(256) void k_bnstats(
    const float* __restrict__ y, float* __restrict__ sums)
{
  __shared__ float s1[256], s2[256];
  const int c = blockIdx.x;
  float a = 0.f, b2 = 0.f;
  for (int t = threadIdx.x; t < NB * NPIX; t += 256) {
    const int bb = t >> 12, p = t & 4095;
    const float val = y[((size_t)bb * CIN + c) * NPIX + p];
    a += val; b2 += val * val;
  }
  s1[threadIdx.x] = a; s2[threadIdx.x] = b2;
  __syncthreads();
  for (int st = 128; st > 0; st >>= 1) {
    if (threadIdx.x < st) {
      s1[threadIdx.x] += s1[threadIdx.x + st];
      s2[threadIdx.x] += s2[threadIdx.x + st];
    }
    __syncthreads();
  }
  if (threadIdx.x == 0) { sums[c] = s1[0]; sums[256 + c] = s2[0]; }
}

// ---------------- 8. BN apply (biased var) + affine + residual ---------------
__global__ __launch_bounds__(256) void k_bnapply(
    const float* __restrict__ y, const float* __restrict__ sums,
    const float* __restrict__ gamma, const float* __restrict__ beta,
    const float* __restrict__ x, float* __restrict__ yn)
{
  const size_t idx = (size_t)blockIdx.x * 256 + threadIdx.x;  // 4*256*4096
  const int c = (int)((idx >> 12) & 255);
  const float mean = sums[c] * (1.0f / 16384.0f);
  const float var  = sums[256 + c] * (1.0f / 16384.0f) - mean * mean;
  const float inv  = rsqrtf(var + 1e-5f);
  yn[idx] = (y[idx] - mean) * inv * gamma[c] + beta[c] + x[idx];
}

// ---------------- 9. pack conv3 weights: wt[tap][ci][co] ---------------------
__global__ __launch_bounds__(256) void k_packw(
    const float* __restrict__ wc3, float* __restrict__ wt)
{
  const size_t idx = (size_t)blockIdx.x * 256 + threadIdx.x;  // 9*256*256
  const int co  = (int)(idx & 255);
  const int ci  = (int)((idx >> 8) & 255);
  const int tap = (int)(idx >> 16);
  wt[idx] = wc3[((size_t)co * CIN + ci) * 9 + tap];
}

// ---------------- 10. 3x3 conv as implicit GEMM (K = 9 taps x 256 ci) --------
__global__ __launch_bounds__(128) void k_conv3(
    const float* __restrict__ yn, const float* __restrict__ wt,
    const float* __restrict__ bc3, float* __restrict__ out)
{
  LANE_IDS();
  const int w0 = wave * 16;            // 4 waves = one full 64-wide row
  const int h  = blockIdx.x;
  const int c0 = blockIdx.y * 16;
  const int b  = blockIdx.z;

  const float* ynb = yn + (size_t)b * CIN * NPIX;
  v8f acc = {};
  for (int dy = -1; dy <= 1; ++dy) {
    const int hh = h + dy;
    if (hh < 0 || hh >= 64) continue;            // uniform per block
    for (int dx = -1; dx <= 1; ++dx) {
      const int col = w0 + mn + dx;
      const bool ok = (col >= 0) && (col < 64);  // zero-pad via value select
      const float* wp = wt + (size_t)((dy + 1) * 3 + (dx + 1)) * 65536
                           + (size_t)(2 * half) * 256 + c0 + mn;
      const float* yp = ynb + (size_t)(2 * half) * NPIX + (size_t)hh * 64 + (ok ? col : 0);
      for (int ci = 0; ci < CIN; ci += 4) {
        v2f a;  a.x = wp[(size_t)ci * 256];  a.y = wp[(size_t)(ci + 1) * 256];
        v2f bb;
        if (ok) { bb.x = yp[(size_t)ci * NPIX]; bb.y = yp[(size_t)(ci + 1) * NPIX]; }
        else    { bb.x = 0.f; bb.y = 0.f; }
        acc = wmma_f32x4(acc, a, bb);
      }
    }
  }
  float* ob = out + (size_t)b * CIN * NPIX;
#pragma unroll
  for (int r = 0; r < 8; ++r) {
    const int c = c0 + r + 8 * half;
    ob[(size_t)c * NPIX + (size_t)h * 64 + w0 + mn] = acc[r] + bc3[c];
  }
}

extern "C" void kernel_launch(void* const* d_in, const int* in_sizes, int n_in,
                              void* d_out, int out_size, void* d_ws, size_t ws_size,
                              hipStream_t stream)
{
  const float* x     = (const float*)d_in[0];
  const float* w_q   = (const float*)d_in[1];
  const float* b_q   = (const float*)d_in[2];
  const float* w_k   = (const float*)d_in[3];
  const float* b_k   = (const float*)d_in[4];
  const float* w_v   = (const float*)d_in[5];
  const float* b_v   = (const float*)d_in[6];
  const float* w_out = (const float*)d_in[7];
  const float* b_out = (const float*)d_in[8];
  const float* gamma = (const float*)d_in[9];
  const float* beta  = (const float*)d_in[10];
  const float* w_c3  = (const float*)d_in[11];
  const float* b_c3  = (const float*)d_in[12];

  float* out   = (float*)d_out;                        // [4,256,64,64]
  float* alpha = out + (size_t)NB * CIN * NPIX;        // [4,4096,4096] (scores -> softmax in place)

  float* ws   = (float*)d_ws;
  float* q    = ws;                    // 4*64*4096
  float* k    = q   + 1048576;
  float* v    = k   + 1048576;
  float* vT   = v   + 1048576;
  float* xh   = vT  + 1048576;         // contiguous [B,N,D] == raw view [B,64,H,W]
  float* y    = xh  + 1048576;         // 4*256*4096
  float* yn   = y   + 4194304;
  float* wt   = yn  + 4194304;         // 9*256*256
  float* sums = wt  + 589824;          // 512

  k_qkv        <<<dim3(64, 12, 4),  128, 0, stream>>>(x, w_q, b_q, w_k, b_k, w_v, b_v, q, k, v);
  k_scores     <<<dim3(64, 256, 4), 128, 0, stream>>>(q, k, alpha);
  k_softmax    <<<dim3(64),         256, 0, stream>>>(alpha);
  k_transpose_v<<<dim3(4096),       256, 0, stream>>>(v, vT);
  k_xhat       <<<dim3(256, 4),     128, 0, stream>>>(alpha, vT, xh);
  k_ygemm      <<<dim3(64, 16, 4),  128, 0, stream>>>(xh, w_out, b_out, y);
  k_bnstats    <<<dim3(256),        256, 0, stream>>>(y, sums);
  k_bnapply    <<<dim3(16384),      256, 0, stream>>>(y, sums, gamma, beta, x, yn);
  k_packw      <<<dim3(2304),       256, 0, stream>>>(w_c3, wt);
  k_conv3      <<<dim3(64, 16, 4),  128, 0, stream>>>(yn, wt, b_c3, out);
}